// CustomLoss_17875653886357
// MI455X (gfx1250) — compile-verified
//
#include <hip/hip_runtime.h>
#include <hip/hip_bf16.h>

// ContrastiveLoss vs. single anchor (last element), summed to a scalar.
// Memory-bound (512 KB total => ~22 ns at 23.3 TB/s): launch-latency bound.
// WMMA f32 16x16x4 is used as the 32-lane reduction engine so the matrix
// pipe carries the live reduction path; ds_swizzle does the half-wave fold.

typedef __attribute__((ext_vector_type(2))) float v2f;
typedef __attribute__((ext_vector_type(8))) float v8f;

#define MARGIN_F 500.0f

// ds_swizzle group-of-32 pattern: xor_mask=0x10 ([14:10]), or=0, and=0x1F
// -> every lane reads lane^16 (SWAPX16).
#define SWIZZLE_SWAPX16 0x401F

// Full 32-lane sum via one V_WMMA_F32_16X16X4_F32 + one ds_swizzle.
// A layout (16x4 f32): lane l supplies A[l%16][l<16 ? 0 : 2] in vgpr0 and
// A[l%16][l<16 ? 1 : 3] in vgpr1. Partial goes in vgpr0, 0 in vgpr1;
// B = all ones. Then D[m][n] = r_m + r_{m+16} for every n.
// C/D layout: lane L (0-15) vgpr j = D[j][L]; lane L+16 vgpr j = D[j+8][L].
// sum(c[0..7]) gives rows 0-7 pair-sums on lanes 0-15 and rows 8-15 on
// lanes 16-31; folding across the half-wave (xor 16) yields the grand total.
__device__ __forceinline__ float wave_sum32_wmma(float r) {
  v2f a;
  a.x = r;
  a.y = 0.0f;
  v2f b;
  b.x = 1.0f;
  b.y = 1.0f;
  v8f c = {};
  c = __builtin_amdgcn_wmma_f32_16x16x4_f32(
      /*neg_a=*/false, a, /*neg_b=*/false, b,
      /*c_mod=*/(short)0, c, /*reuse_a=*/false, /*reuse_b=*/false);
  float u = c[0] + c[1] + c[2] + c[3] + c[4] + c[5] + c[6] + c[7];
  int ui = __builtin_bit_cast(int, u);
  int vi = __builtin_amdgcn_ds_swizzle(ui, SWIZZLE_SWAPX16);
  u += __builtin_bit_cast(float, vi);
  return u;  // grand total, uniform across the wave
}

__device__ __forceinline__ float loss_elem(int lab, int aLab,
                                           float x, float y, float z,
                                           float ax, float ay, float az) {
  const float dx = ax - x;
  const float dy = ay - y;
  const float dz = az - z;
  const float dist = dx * dx + dy * dy + dz * dz;
  // same label -> dist (anchor itself contributes 0); else hinge.
  return (lab == aLab) ? dist : fmaxf(0.0f, MARGIN_F - dist);
}

__global__ void __launch_bounds__(256)
contrastive_partial_kernel(const int* __restrict__ labels,
                           const float* __restrict__ coords,
                           float* __restrict__ partials, int n) {
  // Anchor = last element (uniform address -> scalar loads, L2-resident).
  const int   aLab = labels[n - 1];
  const float ax = coords[3 * (n - 1) + 0];
  const float ay = coords[3 * (n - 1) + 1];
  const float az = coords[3 * (n - 1) + 2];

  const int stride = (int)(gridDim.x * blockDim.x);  // 16384 as launched
  const int tid = (int)(blockIdx.x * blockDim.x + threadIdx.x);

  // Uniform full iterations (SGPR loop bound -> no per-iter EXEC masking),
  // then a guarded tail. EXEC is all-ones again before the WMMA.
  const int full = n / stride;  // 2 for N=32768
  float p = 0.0f;
  int i = tid;
  for (int it = 0; it < full; ++it, i += stride) {
    p += loss_elem(labels[i], aLab, coords[3 * i + 0], coords[3 * i + 1],
                   coords[3 * i + 2], ax, ay, az);
  }
  if (i < n) {
    p += loss_elem(labels[i], aLab, coords[3 * i + 0], coords[3 * i + 1],
                   coords[3 * i + 2], ax, ay, az);
  }

  const float wsum = wave_sum32_wmma(p);

  __shared__ float sdata[8];
  const int wave = (int)threadIdx.x >> 5;
  const int lane = (int)threadIdx.x & 31;
  if (lane == 0) sdata[wave] = wsum;
  __syncthreads();
  if (threadIdx.x == 0) {
    float s = 0.0f;
#pragma unroll
    for (int w = 0; w < 8; ++w) s += sdata[w];
    partials[blockIdx.x] = s;
  }
}

__global__ void __launch_bounds__(32)
contrastive_final_kernel(const float* __restrict__ partials,
                         float* __restrict__ out, int nPartials) {
  // One full wave. 64 partials -> exactly 2 strided loads per lane.
  float r = 0.0f;
  for (int i = (int)threadIdx.x; i < nPartials; i += 32) r += partials[i];
  const float total = wave_sum32_wmma(r);
  if (threadIdx.x == 0) out[0] = total;  // overwrite every call: deterministic
}

extern "C" void kernel_launch(void* const* d_in, const int* in_sizes, int n_in,
                              void* d_out, int out_size, void* d_ws, size_t ws_size,
                              hipStream_t stream) {
  const int*   labels = (const int*)d_in[0];    // batched_labels, int32[N]
  const float* coords = (const float*)d_in[1];  // batched_predicted_coords, f32[N,3]
  float* out = (float*)d_out;                   // scalar f32
  float* partials = (float*)d_ws;               // 64 f32 scratch (256 B)

  const int n = in_sizes[0];
  const int NBLOCKS = 64;    // all 64 partials overwritten each call
  const int NTHREADS = 256;  // 8 waves per block (wave32)

  contrastive_partial_kernel<<<NBLOCKS, NTHREADS, 0, stream>>>(labels, coords,
                                                               partials, n);
  contrastive_final_kernel<<<1, 32, 0, stream>>>(partials, out, NBLOCKS);
}